// GPRGNN_23149873725490
// MI455X (gfx1250) — compile-verified
//
#include <hip/hip_runtime.h>
#include <math.h>

typedef float v2f __attribute__((ext_vector_type(2)));
typedef float v8f __attribute__((ext_vector_type(8)));

__device__ __forceinline__ v8f wmma_f32_k4(v2f a, v2f b, v8f c) {
  // V_WMMA_F32_16X16X4_F32 : D = A(16x4) * B(4x16) + C(16x16)
  return __builtin_amdgcn_wmma_f32_16x16x4_f32(
      /*neg_a=*/false, a, /*neg_b=*/false, b,
      /*c_mod=*/(short)0, c, /*reuse_a=*/false, /*reuse_b=*/false);
}

// ---------------------------------------------------------------------------
// Degree / dinv
// ---------------------------------------------------------------------------
__global__ void deg_init_kernel(float* __restrict__ deg, int n) {
  int i = blockIdx.x * blockDim.x + threadIdx.x;
  if (i < n) deg[i] = 1.0f;  // self loop
}

__global__ void deg_edges_kernel(const long long* __restrict__ dst,
                                 float* __restrict__ deg, long long E) {
  long long e = (long long)blockIdx.x * blockDim.x + threadIdx.x;
  if (e < E) atomicAdd(&deg[dst[e]], 1.0f);
}

__global__ void rsqrt_kernel(float* __restrict__ deg, int n) {
  int i = blockIdx.x * blockDim.x + threadIdx.x;
  if (i < n) deg[i] = rsqrtf(deg[i]);  // deg >= 1 always (self loops)
}

// ---------------------------------------------------------------------------
// GEMM1: hlin[N,16] = x[N,512] @ W1[512,16]  (WMMA f32 16x16x4, 128 k-steps)
// ---------------------------------------------------------------------------
__global__ void gemm1_kernel(const float* __restrict__ x,
                             const float* __restrict__ W1,
                             float* __restrict__ hlin, int ntiles) {
  const int wave = (int)((blockIdx.x * blockDim.x + threadIdx.x) >> 5);
  const int lane = threadIdx.x & 31;
  if (wave >= ntiles) return;  // wave-uniform: EXEC stays all-1 for WMMA
  const int hf = lane >> 4;    // half-wave id
  const int m  = lane & 15;
  const int row0 = wave << 4;

  const float* xrow = x + (size_t)(row0 + m) * 512 + (hf << 1);
  v8f acc = {};
  for (int k = 0; k < 512; k += 4) {
    v2f a = *(const v2f*)(xrow + k);           // A: row m, K = k+2*hf+{0,1}
    const int kb = k + (hf << 1);
    v2f b;
    b.x = W1[kb * 16 + m];                     // B: col m, rows kb, kb+1
    b.y = W1[kb * 16 + 16 + m];
    acc = wmma_f32_k4(a, b, acc);
  }
  float* out = hlin + (size_t)row0 * 16 + m;
#pragma unroll
  for (int r = 0; r < 8; ++r)
    out[(size_t)(r + (hf << 3)) * 16] = acc[r];  // C: row r+8*hf, col m
}

// ---------------------------------------------------------------------------
// Self-loop init + edge scatter-add (16 features)
// ---------------------------------------------------------------------------
__global__ void selfloop16_kernel(const float* __restrict__ hin,
                                  const float* __restrict__ dinv,
                                  float* __restrict__ agg, int n16) {
  int idx = blockIdx.x * blockDim.x + threadIdx.x;
  if (idx >= n16) return;
  float dv = dinv[idx >> 4];
  agg[idx] = hin[idx] * dv * dv;
}

__global__ void agg_edges16_kernel(const long long* __restrict__ src,
                                   const long long* __restrict__ dst,
                                   const float* __restrict__ hin,
                                   const float* __restrict__ dinv,
                                   float* __restrict__ agg, long long E) {
  long long e = (long long)blockIdx.x * blockDim.x + threadIdx.x;
  if (e >= E) return;
  long long s = src[e], d = dst[e];
  float norm = dinv[s] * dinv[d];
  const float4* hr = (const float4*)(hin + s * 16);
  float* ar = agg + d * 16;
#pragma unroll
  for (int q = 0; q < 4; ++q) {
    float4 v = hr[q];
    atomicAdd(ar + 4 * q + 0, v.x * norm);
    atomicAdd(ar + 4 * q + 1, v.y * norm);
    atomicAdd(ar + 4 * q + 2, v.z * norm);
    atomicAdd(ar + 4 * q + 3, v.w * norm);
  }
}

__global__ void relu_bias_kernel(const float* __restrict__ agg,
                                 const float* __restrict__ b1,
                                 float* __restrict__ h, int n16) {
  int idx = blockIdx.x * blockDim.x + threadIdx.x;
  if (idx >= n16) return;
  h[idx] = fmaxf(agg[idx] + b1[idx & 15], 0.0f);
}

// ---------------------------------------------------------------------------
// 10-hop GPR propagation: h = 0.1*relu(h@Wl + bl) + 0.9*h  (per-tile local!)
// One wave owns a 16-row tile; 4 WMMAs per hop, LDS round-trip for D->A.
// ---------------------------------------------------------------------------
#define PROP_WAVES 8
__global__ void prop_kernel(float* __restrict__ h, const float* __restrict__ Wl,
                            const float* __restrict__ bl, int ntiles) {
  __shared__ float tile[PROP_WAVES][16][17];  // padded vs bank conflicts
  const int wib  = threadIdx.x >> 5;  // wave-in-block
  const int wave = (int)((blockIdx.x * blockDim.x + threadIdx.x) >> 5);
  const int lane = threadIdx.x & 31;
  const bool active = wave < ntiles;
  const int hf = lane >> 4;
  const int m  = lane & 15;
  const int row0 = wave << 4;

  if (active) {
#pragma unroll
    for (int j = 0; j < 8; ++j) {
      int idx = lane * 8 + j;  // 0..255 over the 16x16 tile, contiguous in h
      tile[wib][idx >> 4][idx & 15] = h[(size_t)row0 * 16 + idx];
    }
  }
  // Wl B-fragments + bias stay in registers for all 10 hops
  v2f bf[4];
#pragma unroll
  for (int s = 0; s < 4; ++s) {
    int kb = 4 * s + 2 * hf;
    bf[s].x = Wl[kb * 16 + m];
    bf[s].y = Wl[kb * 16 + 16 + m];
  }
  float blm = bl[m];
  __syncthreads();

  for (int it = 0; it < 10; ++it) {
    if (active) {
      v8f acc = {};
#pragma unroll
      for (int s = 0; s < 4; ++s) {
        v2f a;
        a.x = tile[wib][m][4 * s + 2 * hf];
        a.y = tile[wib][m][4 * s + 2 * hf + 1];
        acc = wmma_f32_k4(a, bf[s], acc);
      }
#pragma unroll
      for (int r = 0; r < 8; ++r) {
        int mr = r + (hf << 3);
        float old = tile[wib][mr][m];
        tile[wib][mr][m] = 0.1f * fmaxf(acc[r] + blm, 0.0f) + 0.9f * old;
      }
    }
    __syncthreads();
  }

  if (active) {
#pragma unroll
    for (int j = 0; j < 8; ++j) {
      int idx = lane * 8 + j;
      h[(size_t)row0 * 16 + idx] = tile[wib][idx >> 4][idx & 15];
    }
  }
}

// ---------------------------------------------------------------------------
// GEMM2: hw2[N,7] = h[N,16] @ W2[16,7]  (B padded to 16 cols with zeros)
// ---------------------------------------------------------------------------
__global__ void gemm2_kernel(const float* __restrict__ h,
                             const float* __restrict__ W2,
                             float* __restrict__ hw2, int ntiles) {
  const int wave = (int)((blockIdx.x * blockDim.x + threadIdx.x) >> 5);
  const int lane = threadIdx.x & 31;
  if (wave >= ntiles) return;
  const int hf = lane >> 4;
  const int m  = lane & 15;
  const int row0 = wave << 4;

  const float* hrow = h + (size_t)(row0 + m) * 16 + (hf << 1);
  v8f acc = {};
#pragma unroll
  for (int s = 0; s < 4; ++s) {
    v2f a = *(const v2f*)(hrow + 4 * s);
    const int kb = 4 * s + (hf << 1);
    v2f b;
    b.x = (m < 7) ? W2[kb * 7 + m] : 0.0f;
    b.y = (m < 7) ? W2[(kb + 1) * 7 + m] : 0.0f;
    acc = wmma_f32_k4(a, b, acc);
  }
  if (m < 7) {
#pragma unroll
    for (int r = 0; r < 8; ++r)
      hw2[(size_t)(row0 + r + (hf << 3)) * 7 + m] = acc[r];
  }
}

// ---------------------------------------------------------------------------
// Self-loop init + edge scatter-add (7 features), then log-softmax in place
// ---------------------------------------------------------------------------
__global__ void selfloop7_kernel(const float* __restrict__ hw2,
                                 const float* __restrict__ dinv,
                                 float* __restrict__ out, int n7) {
  int idx = blockIdx.x * blockDim.x + threadIdx.x;
  if (idx >= n7) return;
  float dv = dinv[idx / 7];
  out[idx] = hw2[idx] * dv * dv;
}

__global__ void agg_edges7_kernel(const long long* __restrict__ src,
                                  const long long* __restrict__ dst,
                                  const float* __restrict__ hw2,
                                  const float* __restrict__ dinv,
                                  float* __restrict__ out, long long E) {
  long long e = (long long)blockIdx.x * blockDim.x + threadIdx.x;
  if (e >= E) return;
  long long s = src[e], d = dst[e];
  float norm = dinv[s] * dinv[d];
  const float* hr = hw2 + s * 7;
  float* orow = out + d * 7;
#pragma unroll
  for (int c = 0; c < 7; ++c) atomicAdd(orow + c, hr[c] * norm);
}

__global__ void logsoftmax_kernel(float* __restrict__ out,
                                  const float* __restrict__ b2, int n) {
  int i = blockIdx.x * blockDim.x + threadIdx.x;
  if (i >= n) return;
  float z[7];
  float mx = -INFINITY;
#pragma unroll
  for (int c = 0; c < 7; ++c) {
    z[c] = out[i * 7 + c] + b2[c];
    mx = fmaxf(mx, z[c]);
  }
  float sum = 0.0f;
#pragma unroll
  for (int c = 0; c < 7; ++c) sum += expf(z[c] - mx);
  float lse = mx + logf(sum);
#pragma unroll
  for (int c = 0; c < 7; ++c) out[i * 7 + c] = z[c] - lse;
}

// ---------------------------------------------------------------------------
extern "C" void kernel_launch(void* const* d_in, const int* in_sizes, int n_in,
                              void* d_out, int out_size, void* d_ws, size_t ws_size,
                              hipStream_t stream) {
  const float* x      = (const float*)d_in[0];
  const long long* ei = (const long long*)d_in[1];
  const float* W1     = (const float*)d_in[2];
  const float* b1     = (const float*)d_in[3];
  const float* Wl     = (const float*)d_in[4];
  const float* bl     = (const float*)d_in[5];
  const float* W2     = (const float*)d_in[6];
  const float* b2     = (const float*)d_in[7];

  const int N = in_sizes[0] / 512;          // 100000
  const long long E = in_sizes[1] / 2;      // 3200000
  const long long* src = ei;
  const long long* dst = ei + E;
  const int ntiles = (N + 15) / 16;         // 6250 (N divisible by 16)

  // workspace layout (256B aligned chunks)
  char* ws = (char*)d_ws;
  size_t off = 0;
  auto take = [&](size_t bytes) {
    void* p = ws + off;
    off += (bytes + 255) & ~(size_t)255;
    return p;
  };
  float* dinv = (float*)take((size_t)N * 4);          // deg -> dinv in place
  float* bufA = (float*)take((size_t)N * 16 * 4);     // hlin, then h
  float* bufB = (float*)take((size_t)N * 16 * 4);     // aggregation
  float* hw2  = (float*)take((size_t)N * 7 * 4);
  float* outp = (float*)d_out;

  const int TPB = 256;
  const int nblkN   = (N + TPB - 1) / TPB;
  const int nblkN16 = (N * 16 + TPB - 1) / TPB;
  const int nblkN7  = (N * 7 + TPB - 1) / TPB;
  const int nblkE   = (int)((E + TPB - 1) / TPB);
  const int nblkT   = (ntiles + (TPB / 32) - 1) / (TPB / 32);

  // 1) degrees -> dinv
  deg_init_kernel<<<nblkN, TPB, 0, stream>>>(dinv, N);
  deg_edges_kernel<<<nblkE, TPB, 0, stream>>>(dst, dinv, E);
  rsqrt_kernel<<<nblkN, TPB, 0, stream>>>(dinv, N);

  // 2) conv1: linear (WMMA), aggregate, bias+relu
  gemm1_kernel<<<nblkT, TPB, 0, stream>>>(x, W1, bufA, ntiles);
  selfloop16_kernel<<<nblkN16, TPB, 0, stream>>>(bufA, dinv, bufB, N * 16);
  agg_edges16_kernel<<<nblkE, TPB, 0, stream>>>(src, dst, bufA, dinv, bufB, E);
  relu_bias_kernel<<<nblkN16, TPB, 0, stream>>>(bufB, b1, bufA, N * 16);

  // 3) 10-hop GPR propagation (single kernel, 40 WMMAs/wave)
  prop_kernel<<<nblkT, TPB, 0, stream>>>(bufA, Wl, bl, ntiles);

  // 4) conv2: linear (WMMA), aggregate into d_out, bias + log_softmax
  gemm2_kernel<<<nblkT, TPB, 0, stream>>>(bufA, W2, hw2, ntiles);
  selfloop7_kernel<<<nblkN7, TPB, 0, stream>>>(hw2, dinv, outp, N * 7);
  agg_edges7_kernel<<<nblkE, TPB, 0, stream>>>(src, dst, hw2, dinv, outp, E);
  logsoftmax_kernel<<<nblkN, TPB, 0, stream>>>(outp, b2, N);
}